// RNetC_33088428048695
// MI455X (gfx1250) — compile-verified
//
#include <hip/hip_runtime.h>
#include <math.h>

typedef float v2f __attribute__((ext_vector_type(2)));
typedef float v8f __attribute__((ext_vector_type(8)));

// ---------------- workspace layout (float offsets, all 16B aligned) ----
#define WS_XM    0        // 2056  [x0,x1,x8,x9,x10, h1_s, h2_s, pad]
#define WS_XK1   2056     // 2052  [x3, h1_s, f1_s, pad]
#define WS_XK2   4108     // 2052  [x6, h2_s, f2_s, pad]
#define WS_XF1   6160     // 1028  [x4, k1_s, pad]
#define WS_XF2   7188     // 1028  [x7, k2_s, pad]
#define WS_XH1   8216     // 2052  [x2, out_m[:1024], k1_s, pad]
#define WS_XH2   10268    // 2052  [x5, out_m[1024:], k2_s, pad]
#define WS_MNEW  12320    // 2048
#define WS_OUTM  14368    // 2048
#define WS_H1N   16416    // 1024
#define WS_H2N   17440
#define WS_K1N   18464
#define WS_K2N   19488
#define WS_F1N   20512
#define WS_F2N   21536    // total 22560 floats (~90 KB)

struct GruDesc {
  const float* Wih; const float* Whh; const float* bih; const float* bhh;
  const float* xvec; const float* hstate; float* hout; int xlen;
};
struct GruDescPack { GruDesc d[6]; };

// -----------------------------------------------------------------------
// Partial GEMV-16 over K range [kbeg,kend) via V_WMMA_F32_16X16X4_F32.
// A: 16 rows x 4 K of W (lane m = row m; lane halves hold K pairs, per the
// ISA 32-bit A layout).  B: x[k..k+3] broadcast into all 16 columns via
// LDS same-address reads.  Two accumulators break the D->C chain; uniform
// (SGPR) loop bounds keep EXEC all-ones and let the unroller clause loads.
// Result: lane 0 holds rows R..R+7 in acc[0..7], lane 16 rows R+8..R+15.
// -----------------------------------------------------------------------
__device__ __forceinline__ v8f gemv16_wmma(const float* __restrict__ W, int ld,
                                           int rowbase, const float* vls,
                                           int kbeg, int kend, int lane) {
  v8f acc0 = {0.f, 0.f, 0.f, 0.f, 0.f, 0.f, 0.f, 0.f};
  v8f acc1 = {0.f, 0.f, 0.f, 0.f, 0.f, 0.f, 0.f, 0.f};
  const int boff = (lane >> 4) << 1;  // 0 for lanes 0-15, 2 for lanes 16-31
  const float* wr = W + (size_t)(rowbase + (lane & 15)) * (size_t)ld + boff;
  int k = kbeg;
#pragma unroll 4
  for (; k + 8 <= kend; k += 8) {
    v2f a0; a0.x = wr[k];     a0.y = wr[k + 1];
    v2f a1; a1.x = wr[k + 4]; a1.y = wr[k + 5];
    v2f b0; b0.x = vls[k + boff];     b0.y = vls[k + boff + 1];
    v2f b1; b1.x = vls[k + 4 + boff]; b1.y = vls[k + 4 + boff + 1];
    acc0 = __builtin_amdgcn_wmma_f32_16x16x4_f32(false, a0, false, b0,
                                                 (short)0, acc0, false, false);
    acc1 = __builtin_amdgcn_wmma_f32_16x16x4_f32(false, a1, false, b1,
                                                 (short)0, acc1, false, false);
  }
  if (k < kend) {  // K ranges are 4-aligned; at most one extra 4-block
    v2f a; a.x = wr[k]; a.y = wr[k + 1];
    v2f b; b.x = vls[k + boff]; b.y = vls[k + boff + 1];
    acc0 = __builtin_amdgcn_wmma_f32_16x16x4_f32(false, a, false, b,
                                                 (short)0, acc0, false, false);
  }
#pragma unroll
  for (int r = 0; r < 8; ++r) acc0[r] += acc1[r];
  return acc0;
}

// -----------------------------------------------------------------------
// Fused GRU step.  Block = 384 threads = 12 waves:
//   wave 2g+s   (g=0..2, s=0..1): gate g of Wih@x, K-half s
//   wave 6+2g+s (g=0..2, s=0..1): gate g of Whh@h, K-half s
// Partials meet in LDS; 16 threads fuse the gates.  grid = (H/16, nGRU).
// -----------------------------------------------------------------------
__global__ __launch_bounds__(384) void gru_kernel(GruDescPack p, int H)
{
  __shared__ float xls[2056];
  __shared__ float hls[2048];
  __shared__ float gls[192];

  const GruDesc g = p.d[blockIdx.y];
  const int tid  = threadIdx.x;
  const int lane = tid & 31;
  const int wu   = __builtin_amdgcn_readfirstlane(tid >> 5);  // uniform wave id
  const int blk  = blockIdx.x;

  const int xlen   = g.xlen;
  const int xlen4p = (xlen + 3) & ~3;  // pack kernels zero-padded to this
  {
    const float4* xs = (const float4*)g.xvec;
    const float4* hs = (const float4*)g.hstate;
    for (int i = tid; i < (xlen4p >> 2); i += 384) ((float4*)xls)[i] = xs[i];
    for (int i = tid; i < (H >> 2);      i += 384) ((float4*)hls)[i] = hs[i];
  }
  __syncthreads();

  const bool ih    = wu < 6;
  const float* W   = ih ? g.Wih : g.Whh;
  const float* bb  = ih ? g.bih : g.bhh;
  const float* vls = ih ? xls  : hls;
  const int len    = ih ? xlen : H;
  const int len4   = len & ~3;
  const int gate   = (ih ? wu : (wu - 6)) >> 1;
  const int khalf  = wu & 1;
  const int rowbase = gate * H + blk * 16;
  const int half   = ((len4 >> 1) + 3) & ~3;       // 4-aligned split point
  const int kbeg   = khalf ? half : 0;
  const int kend   = khalf ? len4 : half;

  v8f acc = gemv16_wmma(W, len, rowbase, vls, kbeg, kend, lane);

  if (lane == 0) {
#pragma unroll
    for (int r = 0; r < 8; ++r) gls[wu * 16 + r] = acc[r];
  } else if (lane == 16) {
#pragma unroll
    for (int r = 0; r < 8; ++r) gls[wu * 16 + 8 + r] = acc[r];
  }
  __syncthreads();

  if (lane < 16) {
    const int row = rowbase + lane;
    float gv = gls[wu * 16 + lane];
    if (khalf == 0) {
      gv += bb[row];                                // bias once per gate
    } else {
      for (int t = len4; t < len; ++t)              // ragged tail (len%4==1)
        gv += W[(size_t)row * (size_t)len + t] * vls[t];
    }
    gls[wu * 16 + lane] = gv;
  }
  __syncthreads();

  if (tid < 16) {
    const int i = blk * 16 + tid;
    const float gxr = gls[tid]       + gls[16 + tid];
    const float gxz = gls[32 + tid]  + gls[48 + tid];
    const float gxn = gls[64 + tid]  + gls[80 + tid];
    const float ghr = gls[96 + tid]  + gls[112 + tid];
    const float ghz = gls[128 + tid] + gls[144 + tid];
    const float ghn = gls[160 + tid] + gls[176 + tid];
    const float r = 1.f / (1.f + expf(-(gxr + ghr)));
    const float z = 1.f / (1.f + expf(-(gxz + ghz)));
    const float n = tanhf(gxn + r * ghn);
    g.hout[i] = (1.f - z) * n + z * hls[i];
  }
}

// -----------------------------------------------------------------------
// m_out linear: out = W @ x + b.  Block = 64 threads = 2 waves, each wave
// one K-half of 16 rows; grid = rows/16 = 128 blocks.  len multiple of 8.
// -----------------------------------------------------------------------
__global__ __launch_bounds__(64) void linear16_kernel(
    const float* __restrict__ W, const float* __restrict__ b,
    const float* __restrict__ x, int len, float* __restrict__ out)
{
  __shared__ float xls[2048];
  __shared__ float sls[32];
  const int tid = threadIdx.x, lane = tid & 31;
  const int wu = __builtin_amdgcn_readfirstlane(tid >> 5);
  for (int i = tid; i < (len >> 2); i += 64)
    ((float4*)xls)[i] = ((const float4*)x)[i];
  __syncthreads();

  const int rowbase = blockIdx.x * 16;
  const int half = len >> 1;
  v8f acc = gemv16_wmma(W, len, rowbase, xls, wu * half, wu * half + half, lane);

  if (lane == 0) {
#pragma unroll
    for (int r = 0; r < 8; ++r) sls[wu * 16 + r] = acc[r];
  } else if (lane == 16) {
#pragma unroll
    for (int r = 0; r < 8; ++r) sls[wu * 16 + 8 + r] = acc[r];
  }
  __syncthreads();
  if (tid < 16)
    out[rowbase + tid] = sls[tid] + sls[16 + tid] + b[rowbase + tid];
}

// -----------------------------------------------------------------------
// Pack kernels: build concatenated (zero-padded) GEMV input vectors.
// Stage 0: 5 tasks in one launch (grid.y = task).
// -----------------------------------------------------------------------
__global__ void pack_stage0(const float* __restrict__ x,
                            const float* __restrict__ h1,
                            const float* __restrict__ h2,
                            const float* __restrict__ k1,
                            const float* __restrict__ k2,
                            const float* __restrict__ f1,
                            const float* __restrict__ f2,
                            float* __restrict__ ws)
{
  const int task = blockIdx.y;
  const int i = blockIdx.x * 256 + threadIdx.x;
  if (task == 0) {                      // xm (2056)
    if (i >= 2056) return;
    float v;
    if      (i == 0)   v = x[0];
    else if (i == 1)   v = x[1];
    else if (i == 2)   v = x[8];
    else if (i == 3)   v = x[9];
    else if (i == 4)   v = x[10];
    else if (i < 1029) v = h1[i - 5];
    else if (i < 2053) v = h2[i - 1029];
    else               v = 0.f;
    ws[WS_XM + i] = v;
    return;
  }
  const int   xi[5]  = {0, 3, 6, 4, 7};
  const float* av[5] = {0, h1, h2, k1, k2};
  const float* bv[5] = {0, f1, f2, 0, 0};
  const int   tot[5] = {0, 2052, 2052, 1028, 1028};
  const int   off[5] = {0, WS_XK1, WS_XK2, WS_XF1, WS_XF2};
  const int   bl[5]  = {0, 1024, 1024, 0, 0};
  if (i >= tot[task]) return;
  float v;
  if      (i == 0)                v = x[xi[task]];
  else if (i < 1025)              v = av[task][i - 1];
  else if (i < 1025 + bl[task])   v = bv[task][i - 1025];
  else                            v = 0.f;
  ws[off[task] + i] = v;
}

// Stage 2: xh1/xh2 from out_m (grid.y = task).
__global__ void pack_stage2(const float* __restrict__ x,
                            const float* __restrict__ k1,
                            const float* __restrict__ k2,
                            float* __restrict__ ws)
{
  const int task = blockIdx.y;          // 0 -> xh1, 1 -> xh2
  const int i = blockIdx.x * 256 + threadIdx.x;
  if (i >= 2052) return;
  const float* outm = ws + WS_OUTM + task * 1024;
  const float* kk   = task ? k2 : k1;
  float v;
  if      (i == 0)    v = x[task ? 5 : 2];
  else if (i < 1025)  v = outm[i - 1];
  else if (i < 2049)  v = kk[i - 1025];
  else                v = 0.f;
  ws[(task ? WS_XH2 : WS_XH1) + i] = v;
}

// -----------------------------------------------------------------------
// Output heads: 6 dot products of length 1024 (+bias).  1 block of
// 6 waves; wave32 shuffle reduction.
// -----------------------------------------------------------------------
__global__ __launch_bounds__(192) void heads_kernel(
    const float* __restrict__ h1n, const float* __restrict__ k1n,
    const float* __restrict__ f1n, const float* __restrict__ h2n,
    const float* __restrict__ k2n, const float* __restrict__ f2n,
    const float* __restrict__ w0, const float* __restrict__ b0,
    const float* __restrict__ w1, const float* __restrict__ b1,
    const float* __restrict__ w2, const float* __restrict__ b2,
    const float* __restrict__ w3, const float* __restrict__ b3,
    const float* __restrict__ w4, const float* __restrict__ b4,
    const float* __restrict__ w5, const float* __restrict__ b5,
    float* __restrict__ out)
{
  const int tid = threadIdx.x, lane = tid & 31, w = tid >> 5;
  const float* vecs[6] = {h1n, k1n, f1n, h2n, k2n, f2n};
  const float* wts[6]  = {w0, w1, w2, w3, w4, w5};
  const float* bs[6]   = {b0, b1, b2, b3, b4, b5};
  const float4* v  = (const float4*)vecs[w];
  const float4* ww = (const float4*)wts[w];
  float s = 0.f;
  for (int t = lane; t < 256; t += 32) {
    float4 a = v[t], c = ww[t];
    s += a.x * c.x + a.y * c.y + a.z * c.z + a.w * c.w;
  }
#pragma unroll
  for (int off = 16; off > 0; off >>= 1) s += __shfl_down(s, off, 32);
  if (lane == 0) out[w] = bs[w][0] + s;
}

// -----------------------------------------------------------------------
extern "C" void kernel_launch(void* const* d_in, const int* in_sizes, int n_in,
                              void* d_out, int out_size, void* d_ws, size_t ws_size,
                              hipStream_t stream)
{
  (void)in_sizes; (void)n_in; (void)out_size; (void)ws_size;
  const float* x    = (const float*)d_in[0];
  const float* m_s  = (const float*)d_in[1];
  const float* h1_s = (const float*)d_in[2];
  const float* h2_s = (const float*)d_in[3];
  const float* k1_s = (const float*)d_in[4];
  const float* k2_s = (const float*)d_in[5];
  const float* f1_s = (const float*)d_in[6];
  const float* f2_s = (const float*)d_in[7];
  // params flattened in dict order: per GRU {Wih,Whh,bih,bhh}, per lin {W,b}
  const float* P[42];
  for (int i = 0; i < 42; ++i) P[i] = (const float*)d_in[8 + i];
  const float *mWih = P[0],  *mWhh = P[1],  *mbih = P[2],  *mbhh = P[3];
  const float *moW  = P[4],  *mob  = P[5];
  const float *h1Wih = P[6],  *h1Whh = P[7],  *h1bih = P[8],  *h1bhh = P[9];
  const float *h2Wih = P[10], *h2Whh = P[11], *h2bih = P[12], *h2bhh = P[13];
  const float *ho1W = P[14], *ho1b = P[15], *ho2W = P[16], *ho2b = P[17];
  const float *k1Wih = P[18], *k1Whh = P[19], *k1bih = P[20], *k1bhh = P[21];
  const float *k2Wih = P[22], *k2Whh = P[23], *k2bih = P[24], *k2bhh = P[25];
  const float *ko1W = P[26], *ko1b = P[27], *ko2W = P[28], *ko2b = P[29];
  const float *f1Wih = P[30], *f1Whh = P[31], *f1bih = P[32], *f1bhh = P[33];
  const float *f2Wih = P[34], *f2Whh = P[35], *f2bih = P[36], *f2bhh = P[37];
  const float *fo1W = P[38], *fo1b = P[39], *fo2W = P[40], *fo2b = P[41];

  float* ws  = (float*)d_ws;
  float* out = (float*)d_out;

  // Stage 0: pack all out_m-independent input vectors (one launch).
  pack_stage0<<<dim3(9, 5), 256, 0, stream>>>(x, h1_s, h2_s, k1_s, k2_s,
                                              f1_s, f2_s, ws);

  // Stage 1: master GRU (H=2048, inp=2053), then m_out linear.
  GruDescPack pm;
  pm.d[0] = {mWih, mWhh, mbih, mbhh, ws + WS_XM, m_s, ws + WS_MNEW, 2053};
  for (int i = 1; i < 6; ++i) pm.d[i] = pm.d[0];
  gru_kernel<<<dim3(128, 1), 384, 0, stream>>>(pm, 2048);
  linear16_kernel<<<128, 64, 0, stream>>>(moW, mob, ws + WS_MNEW, 2048,
                                          ws + WS_OUTM);

  // Stage 2: the two out_m-dependent vectors (one launch).
  pack_stage2<<<dim3(9, 2), 256, 0, stream>>>(x, k1_s, k2_s, ws);

  // Stage 3: all six H=1024 GRUs in ONE launch (grid.y = which GRU).
  GruDescPack p6;
  p6.d[0] = {h1Wih, h1Whh, h1bih, h1bhh, ws + WS_XH1, h1_s, ws + WS_H1N, 2049};
  p6.d[1] = {h2Wih, h2Whh, h2bih, h2bhh, ws + WS_XH2, h2_s, ws + WS_H2N, 2049};
  p6.d[2] = {k1Wih, k1Whh, k1bih, k1bhh, ws + WS_XK1, k1_s, ws + WS_K1N, 2049};
  p6.d[3] = {k2Wih, k2Whh, k2bih, k2bhh, ws + WS_XK2, k2_s, ws + WS_K2N, 2049};
  p6.d[4] = {f1Wih, f1Whh, f1bih, f1bhh, ws + WS_XF1, f1_s, ws + WS_F1N, 1025};
  p6.d[5] = {f2Wih, f2Whh, f2bih, f2bhh, ws + WS_XF2, f2_s, ws + WS_F2N, 1025};
  gru_kernel<<<dim3(64, 6), 384, 0, stream>>>(p6, 1024);

  // Stage 4: six scalar output heads, in reference concat order.
  heads_kernel<<<1, 192, 0, stream>>>(ws + WS_H1N, ws + WS_K1N, ws + WS_F1N,
                                      ws + WS_H2N, ws + WS_K2N, ws + WS_F2N,
                                      ho1W, ho1b, ko1W, ko1b, fo1W, fo1b,
                                      ho2W, ho2b, ko2W, ko2b, fo2W, fo2b,
                                      out);
}